// Local_CrossAttention_Layer_5987184410795
// MI455X (gfx1250) — compile-verified
//
#include <hip/hip_runtime.h>
#include <math.h>

// ---------------- problem constants ----------------
#define C_CH   64
#define N_PTS  16384
#define K_NBR  32
#define H_HEAD 8
#define D_HEAD 8
#define NT     4              // points per tile
#define MT     (NT * K_NBR)   // 128 columns (flattened n,k) per tile
#define TPW    8              // tiles processed per (persistent) workgroup

// ---------------- LDS layout (floats; padded strides kill bank conflicts) ----
#define NB_STRIDE 136         // 128 + 8 pad  (TDM pad_interval=128dw, pad_amount=8dw)
#define KV_STRIDE 132
#define WQ_STRIDE 66          // 64 + 2 pad   (TDM pad_interval=64dw,  pad_amount=2dw)

#define OFF_NB0  0
#define SZ_NB    (C_CH * NB_STRIDE)     // 8704
#define OFF_NB1  (OFF_NB0 + SZ_NB)
#define OFF_K    (OFF_NB1 + SZ_NB)
#define SZ_KV    (C_CH * KV_STRIDE)     // 8448
#define OFF_V    (OFF_K + SZ_KV)
#define OFF_WQ   (OFF_V + SZ_KV)
#define SZ_WQ    (C_CH * WQ_STRIDE)     // 4224
#define OFF_P0   (OFF_WQ + SZ_WQ)
#define SZ_P     (C_CH * NT)            // 256
#define OFF_P1   (OFF_P0 + SZ_P)
#define OFF_Q    (OFF_P1 + SZ_P)
#define OFF_O0   (OFF_Q + SZ_P)
#define OFF_O1   (OFF_O0 + SZ_P)
#define SMEM_FLOATS (OFF_O1 + SZ_P)     // 39808 floats = 159232 bytes

typedef __attribute__((ext_vector_type(2))) float v2f;
typedef __attribute__((ext_vector_type(8))) float v8f;
typedef unsigned int u32x4 __attribute__((ext_vector_type(4)));
typedef int i32x4 __attribute__((ext_vector_type(4)));
typedef int i32x8 __attribute__((ext_vector_type(8)));

#if defined(__HIP_DEVICE_COMPILE__) && __has_builtin(__builtin_amdgcn_tensor_load_to_lds) && \
    __has_builtin(__builtin_amdgcn_tensor_store_from_lds) && \
    __has_builtin(__builtin_amdgcn_s_wait_tensorcnt)
#define HAVE_TDM 1
#else
#define HAVE_TDM 0
#endif
#define TDM_6ARG (__clang_major__ >= 23)

#if HAVE_TDM
// Build a 2D-tile Tensor-DMA descriptor (D# groups 0/1 per CDNA5 ISA).
__device__ __forceinline__ void tdm_desc(unsigned lds_byte, const void* gptr,
                                         unsigned tile_w, unsigned tile_h,
                                         unsigned long long row_stride_elems,
                                         unsigned pad_i, unsigned pad_a, unsigned pad_en,
                                         u32x4& g0, i32x8& g1)
{
    unsigned long long ga = (unsigned long long)(uintptr_t)gptr;
    g0.x = 1u;                                            // count=1 (valid user D#)
    g0.y = lds_byte;                                      // lds_addr
    g0.z = (unsigned)ga;                                  // global_addr[31:0]
    g0.w = ((unsigned)(ga >> 32) & 0x01ffffffu)           // global_addr[56:32]
         | (2u << 30);                                    // type=2 ("image")
    unsigned d0 = (2u << 16)                              // data_size = 4 bytes
                | (pad_en << 20)                          // pad_enable
                | (pad_i << 22)                           // pad_interval code
                | (pad_a << 25);                          // pad_amount code
    g1[0] = (int)d0;
    g1[1] = (int)((tile_w & 0xffffu) << 16);              // tensor_dim0.lo16 (== tile_w)
    g1[2] = (int)(((tile_w >> 16) & 0xffffu) | ((tile_h & 0xffffu) << 16)); // dim0.hi|dim1.lo
    g1[3] = (int)(((tile_h >> 16) & 0xffffu) | ((tile_w & 0xffffu) << 16)); // dim1.hi|tile_dim0
    g1[4] = (int)(tile_h & 0xffffu);                      // tile_dim1 (tile_dim2 = 0)
    g1[5] = (int)(unsigned)row_stride_elems;              // tensor_dim0_stride.lo32
    g1[6] = (int)(unsigned)((row_stride_elems >> 32) & 0xffffu);
    g1[7] = 0;
}
__device__ __forceinline__ void tdm_load2d(float* lds, const void* g, unsigned w, unsigned h,
                                           unsigned long long rs, unsigned pi, unsigned pa,
                                           unsigned pe)
{
    u32x4 g0; i32x8 g1;
    tdm_desc((unsigned)(uintptr_t)lds, g, w, h, rs, pi, pa, pe, g0, g1);
    i32x4 z4 = 0; i32x8 z8 = 0; (void)z8;
#if TDM_6ARG
    __builtin_amdgcn_tensor_load_to_lds(g0, g1, z4, z4, z8, 0);
#else
    __builtin_amdgcn_tensor_load_to_lds(g0, g1, z4, z4, 0);
#endif
}
__device__ __forceinline__ void tdm_store2d(const float* lds, void* g, unsigned w, unsigned h,
                                            unsigned long long rs)
{
    u32x4 g0; i32x8 g1;
    tdm_desc((unsigned)(uintptr_t)lds, g, w, h, rs, 0u, 0u, 0u, g0, g1);
    i32x4 z4 = 0; i32x8 z8 = 0; (void)z8;
#if TDM_6ARG
    __builtin_amdgcn_tensor_store_from_lds(g0, g1, z4, z4, z8, 0);
#else
    __builtin_amdgcn_tensor_store_from_lds(g0, g1, z4, z4, 0);
#endif
}
#else
// Fallback: wave0 cooperative copy (synchronous). Keeps the file compiling on
// toolchains without the TDM builtins; never the intended fast path.
__device__ __forceinline__ void fb_load2d(float* lds, const float* g, int w, int h,
                                          size_t rs, int ldsStride)
{
    for (int i = (int)(threadIdx.x & 31); i < w * h; i += 32) {
        int r = i / w, c = i - r * w;
        lds[r * ldsStride + c] = g[(size_t)r * rs + c];
    }
}
__device__ __forceinline__ void fb_store2d(const float* lds, float* g, int w, int h,
                                           size_t rs, int ldsStride)
{
    for (int i = (int)(threadIdx.x & 31); i < w * h; i += 32) {
        int r = i / w, c = i - r * w;
        g[(size_t)r * rs + c] = lds[r * ldsStride + c];
    }
}
#endif

__global__ void __launch_bounds__(256)
lca_fused_kernel(const float* __restrict__ pcd,
                 const float* __restrict__ nbr,
                 const float* __restrict__ Wq,
                 const float* __restrict__ Wk,
                 const float* __restrict__ Wv,
                 float* __restrict__ out)
{
    extern __shared__ float smem[];
    // NOTE: no pointer *arrays* here — LDS-derived constant pointer arrays make
    // lld choke on addrspacecast static initializers. Select with ternaries.
    float* sNb0 = smem + OFF_NB0;
    float* sNb1 = smem + OFF_NB1;
    float* sK   = smem + OFF_K;
    float* sV   = smem + OFF_V;
    float* sWq  = smem + OFF_WQ;
    float* sP0  = smem + OFF_P0;
    float* sP1  = smem + OFF_P1;
    float* sQ   = smem + OFF_Q;
    float* sO0  = smem + OFF_O0;
    float* sO1  = smem + OFF_O1;

    const int t    = threadIdx.x;
    const int lane = t & 31;
    const int wave = t >> 5;
    const int b    = blockIdx.y;
    const int tileBase = blockIdx.x * TPW;               // first point-tile of this WG

    const size_t chanStride = (size_t)N_PTS * K_NBR;     // neighbors channel stride

    // ---- Preload per-wave weight A-fragments into registers (L2-resident) ----
    // A 16x4 f32: lanes 0-15 = rows; VGPR pair = K{0,1} (lanes<16) / K{2,3} (lanes>=16)
    const int row  = lane & 15;
    const int half = lane >> 4;
    const int m0   = (wave & 3) * 16;
    const float* Wrole = (wave < 4) ? Wk : Wv;
    float2 wa[16];
    #pragma unroll
    for (int ks = 0; ks < 16; ++ks)
        wa[ks] = *(const float2*)(Wrole + (m0 + row) * C_CH + ks * 4 + half * 2);

    // ---- Wave0: one-time Wq stage + prefetch of tile 0 (async via TDM) ----
    if (wave == 0) {
        const int n0p0 = tileBase * NT;
        const float* gNb0 = nbr + (size_t)b * C_CH * chanStride + (size_t)n0p0 * K_NBR;
        const float* gP0  = pcd + (size_t)b * C_CH * N_PTS + n0p0;
#if HAVE_TDM
        tdm_load2d(sWq, Wq, C_CH, C_CH, C_CH, 5u, 1u, 1u);          // 64x64, pad ->66
        tdm_load2d(sNb0, gNb0, MT, C_CH, chanStride, 6u, 7u, 1u);   // 64x128, pad ->136
        tdm_load2d(sP0, gP0, NT, C_CH, N_PTS, 0u, 0u, 0u);          // 64x4
#else
        fb_load2d(sWq, Wq, C_CH, C_CH, C_CH, WQ_STRIDE);
        fb_load2d(sNb0, gNb0, MT, C_CH, chanStride, NB_STRIDE);
        fb_load2d(sP0, gP0, NT, C_CH, N_PTS, NT);
#endif
    }

    for (int it = 0; it < TPW; ++it) {
        const int n0p = (tileBase + it) * NT;
        const int cur = it & 1;
        float* nbCur = cur ? sNb1 : sNb0;
        float* pCur  = cur ? sP1  : sP0;
        float* oCur  = cur ? sO1  : sO0;

        if (wave == 0) {
            if (it + 1 < TPW) {                 // prefetch next tile into other buffer
                const int n1 = (tileBase + it + 1) * NT;
                const float* gNb = nbr + (size_t)b * C_CH * chanStride + (size_t)n1 * K_NBR;
                const float* gP  = pcd + (size_t)b * C_CH * N_PTS + n1;
                float* nbNxt = cur ? sNb0 : sNb1;
                float* pNxt  = cur ? sP0  : sP1;
#if HAVE_TDM
                tdm_load2d(nbNxt, gNb, MT, C_CH, chanStride, 6u, 7u, 1u);
                tdm_load2d(pNxt, gP, NT, C_CH, N_PTS, 0u, 0u, 0u);
#else
                fb_load2d(nbNxt, gNb, MT, C_CH, chanStride, NB_STRIDE);
                fb_load2d(pNxt, gP, NT, C_CH, N_PTS, NT);
#endif
            }
#if HAVE_TDM
            // In-order TENSORcnt: leave only {prefetch pair, prev store} outstanding.
            if (it == 0)            __builtin_amdgcn_s_wait_tensorcnt(2);
            else if (it == TPW - 1) __builtin_amdgcn_s_wait_tensorcnt(1);
            else                    __builtin_amdgcn_s_wait_tensorcnt(3);
#endif
        }
        __syncthreads();   // tile `it` resident; prev attention done -> sK/sV reusable

        // ---- Q projection: one output per thread (64 chans x NT points) ----
        {
            const int c = t >> 2, p = t & 3;
            float acc = 0.f;
            #pragma unroll
            for (int cc = 0; cc < C_CH; ++cc)
                acc += sWq[c * WQ_STRIDE + cc] * pCur[cc * NT + p];
            sQ[c * NT + p] = acc;
        }

        // ---- K/V projections via V_WMMA_F32_16X16X4_F32 ----
        float* sOut = (wave < 4) ? sK : sV;
        for (int nt = 0; nt < MT / 16; ++nt) {
            const int n0 = nt * 16;
            v8f acc = {0.f, 0.f, 0.f, 0.f, 0.f, 0.f, 0.f, 0.f};
            #pragma unroll
            for (int ks = 0; ks < 16; ++ks) {
                const int kb = ks * 4 + half * 2;
                v2f a;  a.x  = wa[ks].x;
                        a.y  = wa[ks].y;
                v2f bb; bb.x = nbCur[(kb + 0) * NB_STRIDE + n0 + row];
                        bb.y = nbCur[(kb + 1) * NB_STRIDE + n0 + row];
                acc = __builtin_amdgcn_wmma_f32_16x16x4_f32(false, a, false, bb,
                                                            (short)0, acc, false, false);
            }
            #pragma unroll
            for (int v = 0; v < 8; ++v)
                sOut[(m0 + half * 8 + v) * KV_STRIDE + n0 + row] = acc[v];
        }

        __syncthreads();   // sK/sV/sQ visible

        // ---- Local attention (wave0 only: one thread per (point, head)) ----
        if (t < NT * H_HEAD) {
            const int p = t >> 3;
            const int h = t & 7;

            float qv[D_HEAD];
            #pragma unroll
            for (int d = 0; d < D_HEAD; ++d)
                qv[d] = sQ[(h * D_HEAD + d) * NT + p];

            float e[K_NBR];
            float mx = -3.402823466e38f;
            #pragma unroll
            for (int j = 0; j < K_NBR; ++j) {
                float s = 0.f;
                #pragma unroll
                for (int d = 0; d < D_HEAD; ++d)
                    s += qv[d] * sK[(h * D_HEAD + d) * KV_STRIDE + p * K_NBR + j];
                s *= 0.35355339059327373f;   // 1/sqrt(D)
                e[j] = s;
                mx = fmaxf(mx, s);
            }
            float sum = 0.f;
            #pragma unroll
            for (int j = 0; j < K_NBR; ++j) { e[j] = __expf(e[j] - mx); sum += e[j]; }
            const float inv = 1.f / sum;

            #pragma unroll
            for (int d = 0; d < D_HEAD; ++d) {
                float o = 0.f;
                #pragma unroll
                for (int j = 0; j < K_NBR; ++j)
                    o += e[j] * sV[(h * D_HEAD + d) * KV_STRIDE + p * K_NBR + j];
                oCur[(h * D_HEAD + d) * NT + p] = o * inv;
            }
        }

        // ---- Async output: wave0 TDM-stores the 64xNT result tile ----
        if (wave == 0) {
            float* gOut = out + (size_t)b * C_CH * N_PTS + n0p;
#if HAVE_TDM
            asm volatile("s_wait_dscnt 0" ::: "memory");   // sO writes LDS-visible to TDM
            tdm_store2d(oCur, gOut, NT, C_CH, N_PTS);
#else
            fb_store2d(oCur, gOut, NT, C_CH, N_PTS, NT);
#endif
        }
        // next iteration's top __syncthreads orders sK/sV/sO reuse
    }
}

extern "C" void kernel_launch(void* const* d_in, const int* in_sizes, int n_in,
                              void* d_out, int out_size, void* d_ws, size_t ws_size,
                              hipStream_t stream) {
    const float* pcd = (const float*)d_in[0];
    const float* nbr = (const float*)d_in[1];
    const float* Wq  = (const float*)d_in[2];
    const float* Wk  = (const float*)d_in[3];
    const float* Wv  = (const float*)d_in[4];
    float* out = (float*)d_out;

    const size_t smem = (size_t)SMEM_FLOATS * sizeof(float);   // ~155.5 KB < 320 KB WGP LDS
    (void)hipFuncSetAttribute(reinterpret_cast<const void*>(lca_fused_kernel),
                              hipFuncAttributeMaxDynamicSharedMemorySize, (int)smem);

    dim3 grid((N_PTS / NT) / TPW, 2);   // 512 x B persistent-ish WGs, 8 tiles each
    lca_fused_kernel<<<grid, 256, smem, stream>>>(pcd, nbr, Wq, Wk, Wv, out);
}